// DTCWTLayerBiort_Inv_30253749633563
// MI455X (gfx1250) — compile-verified
//
#include <hip/hip_runtime.h>
#include <hip/hip_bf16.h>

typedef __attribute__((ext_vector_type(2))) float v2f;
typedef __attribute__((ext_vector_type(8))) float v8f;

#define IMG_H 256
#define IMG_W 256
#define HALF  128
#define SC 0.70710678118654752440f

__device__ __forceinline__ int mir(int i, int n) {
    if (i < 0) i = -1 - i;
    else if (i >= n) i = 2 * n - 1 - i;
    return i;
}

// c2q quad value at full-res (r,c) for band pair (offA, offB). Branch-free:
// quad[2r,2c]     = SC*(wr0+wr1)   quad[2r,2c+1]   = SC*(wi0+wi1)
// quad[2r+1,2c]   = SC*(wi0-wi1)   quad[2r+1,2c+1] = SC*(wr1-wr0)
__device__ __forceinline__ float quadval(const float* __restrict__ Yhr,
                                         const float* __restrict__ Yhi,
                                         int offA, int offB, int r, int c) {
    const int rh = r >> 1, ch = c >> 1;
    const int pr = r & 1,  pc = c & 1;
    const int o0 = offA + rh * HALF + ch;
    const int o1 = offB + rh * HALF + ch;
    // real pair when parities match, imag pair otherwise (single cndmask on ptr)
    const float* src = (pr == pc) ? Yhr : Yhi;
    const float a = src[o0];
    const float b = src[o1];
    // sign pattern: sa=-1 iff (pr&pc); sb=-1 iff (pr&!pc)
    const float sa = (pr & pc)        ? -1.f : 1.f;
    const float sb = (pr & (pc ^ 1))  ? -1.f : 1.f;
    return SC * (sa * a + sb * b);
}

__global__ __launch_bounds__(256)
void dtcwt_inv_kernel(const float* __restrict__ Yl,
                      const float* __restrict__ Yhr,
                      const float* __restrict__ Yhi,
                      const float* __restrict__ g0o,
                      const float* __restrict__ g1o,
                      float* __restrict__ out) {
    // Row-filtered fields on 36 extended rows x 64 cols, stride 68
    // (68 % 64 == 4 -> adjacent rows land in different LDS banks for phase 2)
    __shared__ float sY1[36 * 68];
    __shared__ float sY2[36 * 68];
    // Banded column-filter matrices, H[r][m] layout (r = extended row 0..19):
    // H0[r*16+m] = g0[r-m-1] (3-tap), H1[r*16+m] = g1[r-m] (5-tap)
    __shared__ float sH0[20 * 16];
    __shared__ float sH1[20 * 16];

    const int cb = blockIdx.x;   // 0..3   : 64-column block
    const int rb = blockIdx.y;   // 0..7   : 32-row block
    const int bc = blockIdx.z;   // 0..255 : (batch, channel) image

    const int tid = threadIdx.x;

    const float g00 = g0o[0], g01 = g0o[1], g02 = g0o[2];
    const float g10 = g1o[0], g11 = g1o[1], g12 = g1o[2], g13 = g1o[3], g14 = g1o[4];

    // ---------- Fill filter-matrix tables (one-time, cheap) ----------
    for (int i = tid; i < 20 * 16; i += 256) {
        const int r = i >> 4;
        const int mm = i & 15;
        const int t0 = r - mm - 1;
        const int t1 = r - mm;
        float h0 = 0.f;
        if (t0 == 0) h0 = g00; else if (t0 == 1) h0 = g01; else if (t0 == 2) h0 = g02;
        float h1 = 0.f;
        if (t1 == 0) h1 = g10; else if (t1 == 1) h1 = g11; else if (t1 == 2) h1 = g12;
        else if (t1 == 3) h1 = g13; else if (t1 == 4) h1 = g14;
        sH0[i] = h0;
        sH1[i] = h1;
    }

    const float* YlImg = Yl + bc * (IMG_H * IMG_W);
    const int hBase = bc * 6 * (HALF * HALF);
    const int lhA = hBase + 0 * (HALF * HALF), lhB = hBase + 5 * (HALF * HALF);
    const int hlA = hBase + 2 * (HALF * HALF), hlB = hBase + 3 * (HALF * HALF);
    const int hhA = hBase + 1 * (HALF * HALF), hhB = hBase + 4 * (HALF * HALF);

    // ---------- Phase 1: row filters into LDS (2304 = 9*256 elems, no divergence) ----------
    for (int idx = tid; idx < 36 * 64; idx += 256) {
        const int re = idx >> 6;          // extended row 0..35 -> global row rb*32-2+re
        const int cc = idx & 63;          // column within block
        const int gr  = mir(rb * 32 - 2 + re, IMG_H);
        const int gc  = cb * 64 + cc;     // always in [0,255]
        const int cm2 = mir(gc - 2, IMG_W);
        const int cm1 = mir(gc - 1, IMG_W);
        const int cp1 = mir(gc + 1, IMG_W);
        const int cp2 = mir(gc + 2, IMG_W);

        const float* row = YlImg + gr * IMG_W;
        float y1 = g00 * row[cm1] + g01 * row[gc] + g02 * row[cp1];
        y1 += g10 * quadval(Yhr, Yhi, hlA, hlB, gr, cm2)
            + g11 * quadval(Yhr, Yhi, hlA, hlB, gr, cm1)
            + g12 * quadval(Yhr, Yhi, hlA, hlB, gr, gc)
            + g13 * quadval(Yhr, Yhi, hlA, hlB, gr, cp1)
            + g14 * quadval(Yhr, Yhi, hlA, hlB, gr, cp2);

        float y2 = g00 * quadval(Yhr, Yhi, lhA, lhB, gr, cm1)
                 + g01 * quadval(Yhr, Yhi, lhA, lhB, gr, gc)
                 + g02 * quadval(Yhr, Yhi, lhA, lhB, gr, cp1);
        y2 += g10 * quadval(Yhr, Yhi, hhA, hhB, gr, cm2)
            + g11 * quadval(Yhr, Yhi, hhA, hhB, gr, cm1)
            + g12 * quadval(Yhr, Yhi, hhA, hhB, gr, gc)
            + g13 * quadval(Yhr, Yhi, hhA, hhB, gr, cp1)
            + g14 * quadval(Yhr, Yhi, hhA, hhB, gr, cp2);

        sY1[re * 68 + cc] = y1;
        sY2[re * 68 + cc] = y2;
    }

    __syncthreads();

    // ---------- Phase 2: column filter via V_WMMA_F32_16X16X4_F32 ----------
    // Out(16x16) = H0(16x20) * Y1ext(20x16) + H1(16x20) * Y2ext(20x16)
    // Straight-line: only ds_load + wmma here (EXEC stays all-1s).
    const int wave = tid >> 5;
    const int lane = tid & 31;
    const int ty = wave >> 2;     // 0..1 tile row
    const int tx = wave & 3;      // 0..3 tile col
    const int m  = lane & 15;     // A: row M; B: col N; D: col N
    const int hi = lane >> 4;     // lane-half selects K/M offset

    v8f acc = {};
    #pragma unroll
    for (int chk = 0; chk < 5; ++chk) {
        const int r0 = 4 * chk + 2 * hi;   // K index for A/B VGPR 0
        const int r1 = r0 + 1;             // K index for A/B VGPR 1
        v2f a0, a1, b1, b2;

        a0.x = sH0[r0 * 16 + m];
        a0.y = sH0[r1 * 16 + m];
        a1.x = sH1[r0 * 16 + m];
        a1.y = sH1[r1 * 16 + m];

        b1.x = sY1[(ty * 16 + r0) * 68 + tx * 16 + m];
        b1.y = sY1[(ty * 16 + r1) * 68 + tx * 16 + m];
        b2.x = sY2[(ty * 16 + r0) * 68 + tx * 16 + m];
        b2.y = sY2[(ty * 16 + r1) * 68 + tx * 16 + m];

        acc = __builtin_amdgcn_wmma_f32_16x16x4_f32(false, a0, false, b1,
                                                    (short)0, acc, false, false);
        acc = __builtin_amdgcn_wmma_f32_16x16x4_f32(false, a1, false, b2,
                                                    (short)0, acc, false, false);
    }

    // D layout: VGPR v -> row (v + 8*hi), col (lane & 15)
    float* outImg = out + bc * (IMG_H * IMG_W);
    const int gcol = cb * 64 + tx * 16 + m;
    #pragma unroll
    for (int v = 0; v < 8; ++v) {
        const int grow = rb * 32 + ty * 16 + v + 8 * hi;
        outImg[grow * IMG_W + gcol] = acc[v];
    }
}

extern "C" void kernel_launch(void* const* d_in, const int* in_sizes, int n_in,
                              void* d_out, int out_size, void* d_ws, size_t ws_size,
                              hipStream_t stream) {
    (void)in_sizes; (void)n_in; (void)out_size; (void)d_ws; (void)ws_size;
    const float* Yl  = (const float*)d_in[0];
    const float* Yhr = (const float*)d_in[1];
    const float* Yhi = (const float*)d_in[2];
    const float* g0o = (const float*)d_in[3];
    const float* g1o = (const float*)d_in[4];
    float* out = (float*)d_out;

    dim3 grid(4, 8, 256);   // col-blocks, row-blocks, b*c images
    dtcwt_inv_kernel<<<grid, 256, 0, stream>>>(Yl, Yhr, Yhi, g0o, g1o, out);
}